// Masking_14654428414188
// MI455X (gfx1250) — compile-verified
//
#include <hip/hip_runtime.h>
#include <hip/hip_bf16.h>
#include <stdint.h>

// ---------------------------------------------------------------------------
// Bottom-k masking via 2x16-bit radix select + exact index-ordered tie break.
// Memory-bound: ~1.5GB total traffic -> ~65us @ 23.3 TB/s on MI455X.
// CDNA5 paths: global_load_async_to_lds_b128 double-buffered pipeline with
// s_wait_asynccnt; global_prefetch_b8 on streaming loops; NT temporal hints
// on single-use streams (weight/out) so the 4x-reused mask stays in L2.
// ---------------------------------------------------------------------------

#ifndef __HIP_DEVICE_COMPILE__
#define USE_ASYNC_LDS 0
#else
#if __has_builtin(__builtin_amdgcn_global_load_async_to_lds_b128) && \
    __has_builtin(__builtin_amdgcn_s_wait_asynccnt)
#define USE_ASYNC_LDS 1
#else
#define USE_ASYNC_LDS 0
#endif
#endif

#if USE_ASYNC_LDS
// Builtin signature (from hipcc diagnostic): (v4i AS1*, v4i AS3*, Ii, Ii)
typedef int mk_v4i __attribute__((vector_size(16)));
typedef mk_v4i __attribute__((address_space(1))) * mk_g4i;  // global, 64-bit
typedef mk_v4i __attribute__((address_space(3))) * mk_l4i;  // LDS, 32-bit
// Generic->AS(1): identical address values on AMDGPU.
#define MK_GPTR(p) ((mk_g4i)(uintptr_t)(p))
// Generic shared->AS(3): low 32 bits of a generic LDS address are the offset.
#define MK_LPTR(p) ((mk_l4i)(unsigned)(uintptr_t)(p))
#endif

// Native clang vector (required by __builtin_nontemporal_*; layout == float4)
typedef float mk_v4f __attribute__((ext_vector_type(4)));

#define HIST_BINS 65536u
#define SC_OFF (2u * HIST_BINS)   // scalars at ws[131072]
#define BT_OFF (SC_OFF + 16u)     // per-block tie counts at ws[131088]
// scalars: [0]=k [1]=bin_hi [2]=rank_in_bin [3]=threshold_key
//          [4]=tie_budget [5]=cum_before_hi

#define MBT 32                    // one wave32 per block in tie/mask kernels
#define MBV 4                     // float4 per lane per iteration
#define MBI 64                    // iterations per block
#define MB_EPB (MBT * MBV * MBI)  // 8192 elements per block

__device__ __forceinline__ unsigned key_of(float f) {
  unsigned u = __float_as_uint(f);
  // monotonic map: ascending key order == ascending float order
  return (u & 0x80000000u) ? ~u : (u | 0x80000000u);
}

// ---------------------------------------------------------------- K0: zero ws
__global__ void mk_zero_kernel(unsigned* ws, unsigned count) {
  unsigned i = blockIdx.x * blockDim.x + threadIdx.x;
  if (i < count) ws[i] = 0u;
}

// ------------------------------------------------------- K1: k from device p
__global__ void mk_compute_k_kernel(const float* p, unsigned n, unsigned* sc) {
  double kp = (1.0 - (double)p[0]) * (double)n;  // matches python int((1-p)*n)
  if (kp < 0.0) kp = 0.0;
  if (kp > (double)n) kp = (double)n;
  sc[0] = (unsigned)kp;  // truncation toward zero, same as int()
}

// ----------------------------------------------- K2: histogram of top 16 bits
// Double-buffered async global->LDS pipeline: issue tile i+1, wait for tile i
// (s_wait_asynccnt 1: async global loads complete in order), consume tile i.
__global__ __launch_bounds__(256) void mk_hist_hi_kernel(const float* m,
                                                         unsigned n,
                                                         unsigned* h1) {
  unsigned i = blockIdx.x * blockDim.x + threadIdx.x;
  unsigned stride = gridDim.x * blockDim.x;
  unsigned n4 = n >> 2;
  const float4* m4 = (const float4*)m;
#if USE_ASYNC_LDS
  __shared__ float4 stage[2][256];
  unsigned cur = 0u;
  if (i < n4) {
    __builtin_amdgcn_global_load_async_to_lds_b128(
        MK_GPTR(&m4[i]), MK_LPTR(&stage[0][threadIdx.x]), 0, 0);
  }
  for (unsigned v = i; v < n4; v += stride) {
    unsigned vn = v + stride;
    if (vn < n4) {
      __builtin_amdgcn_global_load_async_to_lds_b128(
          MK_GPTR(&m4[vn]), MK_LPTR(&stage[cur ^ 1u][threadIdx.x]), 0, 0);
      __builtin_amdgcn_s_wait_asynccnt(1);  // oldest (current tile) done
    } else {
      __builtin_amdgcn_s_wait_asynccnt(0);  // drain
    }
    float4 f = stage[cur][threadIdx.x];
    atomicAdd(&h1[key_of(f.x) >> 16], 1u);
    atomicAdd(&h1[key_of(f.y) >> 16], 1u);
    atomicAdd(&h1[key_of(f.z) >> 16], 1u);
    atomicAdd(&h1[key_of(f.w) >> 16], 1u);
    cur ^= 1u;
  }
  __builtin_amdgcn_s_wait_asynccnt(0);
#else
  for (unsigned v = i; v < n4; v += stride) {
    if (v + stride < n4) __builtin_prefetch(&m4[v + stride], 0, 0);
    float4 f = m4[v];
    atomicAdd(&h1[key_of(f.x) >> 16], 1u);
    atomicAdd(&h1[key_of(f.y) >> 16], 1u);
    atomicAdd(&h1[key_of(f.z) >> 16], 1u);
    atomicAdd(&h1[key_of(f.w) >> 16], 1u);
  }
#endif
  if (i == 0) {
    for (unsigned j = n4 << 2; j < n; ++j)
      atomicAdd(&h1[key_of(m[j]) >> 16], 1u);
  }
}

// -------------------------------- K4: histogram of low 16 bits (bin-filtered)
__global__ __launch_bounds__(256) void mk_hist_lo_kernel(const float* m,
                                                         unsigned n,
                                                         unsigned* h2,
                                                         const unsigned* sc) {
  unsigned bin_hi = sc[1];
  unsigned i = blockIdx.x * blockDim.x + threadIdx.x;
  unsigned stride = gridDim.x * blockDim.x;
  unsigned n4 = n >> 2;
  const float4* m4 = (const float4*)m;
  for (unsigned v = i; v < n4; v += stride) {
    if (v + stride < n4) __builtin_prefetch(&m4[v + stride], 0, 0);
    float4 f = m4[v];
    unsigned k0 = key_of(f.x), k1 = key_of(f.y), k2 = key_of(f.z),
             k3 = key_of(f.w);
    if ((k0 >> 16) == bin_hi) atomicAdd(&h2[k0 & 0xFFFFu], 1u);
    if ((k1 >> 16) == bin_hi) atomicAdd(&h2[k1 & 0xFFFFu], 1u);
    if ((k2 >> 16) == bin_hi) atomicAdd(&h2[k2 & 0xFFFFu], 1u);
    if ((k3 >> 16) == bin_hi) atomicAdd(&h2[k3 & 0xFFFFu], 1u);
  }
  if (i == 0) {
    for (unsigned j = n4 << 2; j < n; ++j) {
      unsigned kk = key_of(m[j]);
      if ((kk >> 16) == bin_hi) atomicAdd(&h2[kk & 0xFFFFu], 1u);
    }
  }
}

// ------------------- K3/K5: single-block scan to locate rank-(k-1) inside hist
__global__ __launch_bounds__(1024) void mk_find_bin_kernel(const unsigned* hist,
                                                           unsigned* sc,
                                                           int phase) {
  __shared__ unsigned part[1024];
  unsigned t = threadIdx.x;
  unsigned k = sc[0];
  if (k == 0u) {  // nothing masked: threshold=0, budget=0
    if (t == 0) {
      if (phase == 0) { sc[1] = 0u; sc[2] = 0u; sc[5] = 0u; }
      else            { sc[3] = 0u; sc[4] = 0u; }
    }
    return;
  }
  unsigned r = (phase == 0) ? (k - 1u) : sc[2];
  unsigned baseb = t * 64u;
  unsigned sum = 0u;
  for (unsigned j = 0; j < 64u; ++j) sum += hist[baseb + j];
  part[t] = sum;
  __syncthreads();
  // Hillis-Steele inclusive scan over 1024 partials
  for (unsigned off = 1u; off < 1024u; off <<= 1) {
    unsigned v = (t >= off) ? part[t - off] : 0u;
    __syncthreads();
    part[t] += v;
    __syncthreads();
  }
  unsigned incl = part[t];
  unsigned excl = incl - sum;
  if (r >= excl && r < incl) {  // exactly one thread matches
    unsigned c = excl, bin = baseb, cumb = excl;
    for (unsigned j = 0; j < 64u; ++j) {
      unsigned h = hist[baseb + j];
      if (r < c + h) { bin = baseb + j; cumb = c; break; }
      c += h;
    }
    if (phase == 0) {
      sc[1] = bin;        // top-16 bin containing rank k-1
      sc[2] = r - cumb;   // residual rank inside that bin
      sc[5] = cumb;       // elements strictly below the bin
    } else {
      unsigned thr = (sc[1] << 16) | bin;
      unsigned less = sc[5] + cumb;  // count(key < threshold)
      sc[3] = thr;
      sc[4] = k - less;              // ties-to-zero budget (index-ordered)
    }
  }
}

// --------------------------------------- K6: per-block count of threshold ties
__global__ __launch_bounds__(MBT) void mk_tie_count_kernel(const float* m,
                                                           unsigned n,
                                                           const unsigned* sc,
                                                           unsigned* bt) {
  unsigned T = sc[3];
  unsigned lane = threadIdx.x;
  unsigned base = blockIdx.x * MB_EPB;
  unsigned total = 0u;
  for (unsigned it = 0; it < MBI; ++it) {
    unsigned i0 = base + it * (MBT * MBV) + lane * MBV;
    __builtin_prefetch(m + i0 + MBT * MBV, 0, 0);
    bool tie[MBV];
    if (i0 + 3u < n) {
      float4 f = *(const float4*)(m + i0);
      tie[0] = key_of(f.x) == T; tie[1] = key_of(f.y) == T;
      tie[2] = key_of(f.z) == T; tie[3] = key_of(f.w) == T;
    } else {
      for (unsigned j = 0; j < MBV; ++j) {
        unsigned idx = i0 + j;
        tie[j] = (idx < n) && (key_of(m[idx]) == T);
      }
    }
    for (unsigned j = 0; j < MBV; ++j)
      total += __popc((unsigned)__ballot((int)tie[j]));
  }
  if (lane == 0) bt[blockIdx.x] = total;
}

// ------------------------------------ K7: exclusive scan of block tie counts
__global__ __launch_bounds__(1024) void mk_tie_scan_kernel(unsigned* bt,
                                                           unsigned nb) {
  __shared__ unsigned part[1024];
  unsigned t = threadIdx.x;
  unsigned chunk = (nb + 1023u) / 1024u;
  unsigned beg = t * chunk;
  unsigned end = beg + chunk; if (end > nb) end = nb;
  unsigned s = 0u;
  for (unsigned i = beg; i < end; ++i) s += bt[i];
  part[t] = s;
  __syncthreads();
  for (unsigned off = 1u; off < 1024u; off <<= 1) {
    unsigned v = (t >= off) ? part[t - off] : 0u;
    __syncthreads();
    part[t] += v;
    __syncthreads();
  }
  unsigned run = part[t] - s;  // exclusive
  for (unsigned i = beg; i < end; ++i) {
    unsigned v = bt[i];
    bt[i] = run;
    run += v;
  }
}

// ------------- K8: fused masked copy with exact index-ordered tie resolution
// weight is read once and out written once -> NT hints keep mask lines in L2.
__global__ __launch_bounds__(MBT) void mk_mask_write_kernel(
    const float* w, const float* m, float* out, unsigned n,
    const unsigned* sc, const unsigned* bt) {
  unsigned T = sc[3];
  unsigned budget = sc[4];
  unsigned lane = threadIdx.x;
  unsigned ltm = (1u << lane) - 1u;
  unsigned base = blockIdx.x * MB_EPB;
  unsigned running = bt[blockIdx.x];  // global tie rank before this block
  for (unsigned it = 0; it < MBI; ++it) {
    unsigned i0 = base + it * (MBT * MBV) + lane * MBV;
    __builtin_prefetch(m + i0 + MBT * MBV, 0, 0);
    float wv[MBV];
    unsigned kv[MBV];
    bool inr[MBV];
    bool vec = (i0 + 3u < n);
    if (vec) {
      mk_v4f wf = __builtin_nontemporal_load((const mk_v4f*)(w + i0));
      float4 mf = *(const float4*)(m + i0);
      wv[0] = wf.x; wv[1] = wf.y; wv[2] = wf.z; wv[3] = wf.w;
      kv[0] = key_of(mf.x); kv[1] = key_of(mf.y);
      kv[2] = key_of(mf.z); kv[3] = key_of(mf.w);
      inr[0] = inr[1] = inr[2] = inr[3] = true;
    } else {
      for (unsigned j = 0; j < MBV; ++j) {
        unsigned idx = i0 + j;
        if (idx < n) { wv[j] = w[idx]; kv[j] = key_of(m[idx]); inr[j] = true; }
        else         { wv[j] = 0.f;   kv[j] = 0u;             inr[j] = false; }
      }
    }
    bool tie[MBV];
    unsigned ballots[MBV];
    unsigned lowsum = 0u, totsum = 0u;
    for (unsigned j = 0; j < MBV; ++j) {
      tie[j] = inr[j] && (kv[j] == T);
      ballots[j] = (unsigned)__ballot((int)tie[j]);  // uniform across wave32
      lowsum += __popc(ballots[j] & ltm);  // ties at lower global indices
      totsum += __popc(ballots[j]);
    }
    unsigned ownpre = 0u;
    float ov[MBV];
    for (unsigned j = 0; j < MBV; ++j) {
      bool z = (kv[j] < T) ||
               (tie[j] && (running + lowsum + ownpre < budget));
      ov[j] = z ? 0.0f : wv[j];
      if (tie[j]) ++ownpre;
    }
    if (vec) {
      mk_v4f of;
      of.x = ov[0]; of.y = ov[1]; of.z = ov[2]; of.w = ov[3];
      __builtin_nontemporal_store(of, (mk_v4f*)(out + i0));
    } else {
      for (unsigned j = 0; j < MBV; ++j) {
        unsigned idx = i0 + j;
        if (idx < n) __builtin_nontemporal_store(ov[j], out + idx);
      }
    }
    running += totsum;  // uniform across the wave
  }
}

// ---------------------------------------------------------------------------
extern "C" void kernel_launch(void* const* d_in, const int* in_sizes, int n_in,
                              void* d_out, int out_size, void* d_ws,
                              size_t ws_size, hipStream_t stream) {
  const float* w = (const float*)d_in[0];
  const float* m = (const float*)d_in[1];
  const float* p = (const float*)d_in[2];  // masking_percent (device scalar)
  float* out = (float*)d_out;
  unsigned n = (unsigned)in_sizes[0];

  unsigned* ws = (unsigned*)d_ws;
  unsigned* h1 = ws;                 // 65536 bins (top 16 bits)
  unsigned* h2 = ws + HIST_BINS;     // 65536 bins (low 16 bits)
  unsigned* sc = ws + SC_OFF;        // scalars
  unsigned* bt = ws + BT_OFF;        // per-block tie counts
  unsigned nb = (n + MB_EPB - 1u) / MB_EPB;

  mk_zero_kernel<<<(BT_OFF + 255u) / 256u, 256, 0, stream>>>(ws, BT_OFF);
  mk_compute_k_kernel<<<1, 1, 0, stream>>>(p, n, sc);
  mk_hist_hi_kernel<<<4096, 256, 0, stream>>>(m, n, h1);
  mk_find_bin_kernel<<<1, 1024, 0, stream>>>(h1, sc, 0);
  mk_hist_lo_kernel<<<4096, 256, 0, stream>>>(m, n, h2, sc);
  mk_find_bin_kernel<<<1, 1024, 0, stream>>>(h2, sc, 1);
  mk_tie_count_kernel<<<nb, MBT, 0, stream>>>(m, n, sc, bt);
  mk_tie_scan_kernel<<<1, 1024, 0, stream>>>(bt, nb);
  mk_mask_write_kernel<<<nb, MBT, 0, stream>>>(w, m, out, n, sc, bt);
}